// GOGraphConv_24215025615565
// MI455X (gfx1250) — compile-verified
//
#include <hip/hip_runtime.h>

#define N_NODES 50000
#define N_EDGES 640000
#define CH      128   // IN_CH == OUT_CH == 128

typedef float v2f __attribute__((ext_vector_type(2)));
typedef float v8f __attribute__((ext_vector_type(8)));

// ---------------------------------------------------------------------------
// Workspace layout (floats): [0, N_NODES) = deg, [N_NODES, N_NODES + N_NODES*CH) = agg/h
// ---------------------------------------------------------------------------

__global__ void gc_init_ws(float* __restrict__ ws, int n) {
    int i = blockIdx.x * blockDim.x + threadIdx.x;
    if (i < n) ws[i] = 0.0f;
}

// One wave32 per edge: gather x[col] as float4 per lane (coalesced b128 loads),
// scatter-add into agg[row] with non-returning f32 atomics (L2-resident target).
__global__ __launch_bounds__(256) void gc_scatter(const float* __restrict__ x,
                                                  const int*   __restrict__ rows,
                                                  const int*   __restrict__ cols,
                                                  float*       __restrict__ deg,
                                                  float*       __restrict__ agg) {
    const int wave = (blockIdx.x * blockDim.x + threadIdx.x) >> 5;  // exact: 640000 waves
    const int lane = threadIdx.x & 31;
    const int r = rows[wave];
    const int c = cols[wave];
    const float4 v = ((const float4*)(x + (size_t)c * CH))[lane];
    float* dst = agg + (size_t)r * CH + lane * 4;
    atomicAdd(dst + 0, v.x);
    atomicAdd(dst + 1, v.y);
    atomicAdd(dst + 2, v.z);
    atomicAdd(dst + 3, v.w);
    if (lane == 0) atomicAdd(deg + r, 1.0f);
}

// h = x + agg / max(deg, 1), written back in place over agg.
__global__ __launch_bounds__(256) void gc_finalize(const float* __restrict__ x,
                                                   const float* __restrict__ deg,
                                                   float*       __restrict__ agg) {
    const int i = blockIdx.x * blockDim.x + threadIdx.x;  // exact: 6.4M threads
    const int node = i >> 7;
    float d = deg[node];
    d = d < 1.0f ? 1.0f : d;
    agg[i] = x[i] + agg[i] / d;
}

// out = h @ W^T + b using V_WMMA_F32_16X16X4_F32.
// One wave per 16x16 output tile; 3125 M-tiles x 8 N-tiles = 25000 waves exactly.
// A frag (16x4 f32, 2 VGPRs/lane): lane<16 -> {K0,K1}, lane>=16 -> {K2,K3}, M = lane%16.
// B frag (4x16 f32): symmetric; N = lane%16, B[k][n] = W[n][k] -> contiguous float2 of W row.
// D/C (16x16 f32, 8 VGPRs): VGPR r -> M = r + 8*(lane/16), N = lane%16.
__global__ __launch_bounds__(256) void gc_gemm_wmma(const float* __restrict__ h,
                                                    const float* __restrict__ W,
                                                    const float* __restrict__ bias,
                                                    float*       __restrict__ out) {
    const int wid  = (blockIdx.x * blockDim.x + threadIdx.x) >> 5;  // 0..24999
    const int lane = threadIdx.x & 31;
    const int mt = wid >> 3;            // 0..3124
    const int nt = wid & 7;             // 0..7
    const int m0 = mt * 16;
    const int n0 = nt * 16;
    const int ln = lane & 15;
    const int hi = lane >> 4;           // 0 or 1

    // Accumulator initialized with bias (N depends only on lane -> same for all 8 VGPRs)
    const float bv = bias[n0 + ln];
    v8f c;
#pragma unroll
    for (int r = 0; r < 8; ++r) c[r] = bv;

    const float* hrow = h + (size_t)(m0 + ln) * CH;   // A row  m = m0 + ln
    const float* wrow = W + (size_t)(n0 + ln) * CH;   // B col  n = n0 + ln  (W[n][k])

#pragma unroll 4
    for (int kb = 0; kb < 32; ++kb) {
        const int k = kb * 4 + hi * 2;                // lane-half selects K pair
        v2f a = *(const v2f*)(hrow + k);
        v2f b = *(const v2f*)(wrow + k);
        // (neg_a, A, neg_b, B, c_mod, C, reuse_a, reuse_b)
        c = __builtin_amdgcn_wmma_f32_16x16x4_f32(false, a, false, b,
                                                  (short)0, c, false, false);
    }

    float* orow = out + (size_t)(m0 + hi * 8) * CH + n0 + ln;
#pragma unroll
    for (int r = 0; r < 8; ++r) orow[(size_t)r * CH] = c[r];
}

// ---------------------------------------------------------------------------

extern "C" void kernel_launch(void* const* d_in, const int* in_sizes, int n_in,
                              void* d_out, int out_size, void* d_ws, size_t ws_size,
                              hipStream_t stream) {
    const float* x    = (const float*)d_in[0];
    const int*   ei   = (const int*)  d_in[1];   // [2, N_EDGES] row-major
    const float* W    = (const float*)d_in[2];   // [OUT_CH, IN_CH]
    const float* bias = (const float*)d_in[3];
    float*       out  = (float*)d_out;

    const int* rows = ei;             // edge_index[0]
    const int* cols = ei + N_EDGES;   // edge_index[1]

    float* deg = (float*)d_ws;
    float* agg = deg + N_NODES;

    // 1) zero deg + agg (must re-zero every call: graph replay keeps ws state)
    {
        const int n = N_NODES + N_NODES * CH;             // 6,450,000
        gc_init_ws<<<(n + 255) / 256, 256, 0, stream>>>((float*)d_ws, n);
    }
    // 2) fused degree + scatter-add: one wave per edge
    {
        const int blocks = (N_EDGES * 32) / 256;          // 80,000 (exact)
        gc_scatter<<<blocks, 256, 0, stream>>>(x, rows, cols, deg, agg);
    }
    // 3) h = x + agg/deg (in place over agg)
    {
        const int blocks = (N_NODES * CH) / 256;          // 25,000 (exact)
        gc_finalize<<<blocks, 256, 0, stream>>>(x, deg, agg);
    }
    // 4) out = h @ W^T + b via f32 WMMA
    {
        const int blocks = (3125 * 8 * 32) / 256;         // 3,125 (exact)
        gc_gemm_wmma<<<blocks, 256, 0, stream>>>(agg, W, bias, out);
    }
}